// VQCStronglyEntanglingModel_10230612099656
// MI455X (gfx1250) — compile-verified
//
#include <hip/hip_runtime.h>

// ---------------------------------------------------------------------------
// VQC StronglyEntanglingLayers simulator for gfx1250 (MI455X).
//  * initial RY-encoded product state = pHi[32] x pLo[32] rank-1 product
//  * per-layer Rot gates fused via Kronecker product into three 16x16
//    complex operators, ALL applied with V_WMMA_F32_16X16X4_F32:
//      G0 = u0 x u1 x u2 x u3   on state bits 9..6   (cols = bits 5..0)
//      G1 = u4 x u5 x u6 x u7   on state bits 5..2   (cols = bits 9..6,1..0)
//      GT = I4  x (u8 x u9)     on state bits 3..0   (cols = bits 9..4)
//    complex = 4 real WMMAs; K=16 contraction = 4 chained K=4 steps.
//    All LDS addressing strength-reduced to hoisted per-lane bases +
//    literal DS offsets.
//  * CNOT-ring permutation (GF(2)-linear) folded into GT's writeback
//    scatter: D(c*16+v+hi8) = table[l][c] ^ D(hi8) ^ D(v), latter literal.
//  * ping-pong state buffers -> exactly one barrier per phase (3/layer)
//  * <Z> on wire 9 (= index LSB) via wave32 shuffle reduction
// ---------------------------------------------------------------------------

typedef float v2f __attribute__((ext_vector_type(2)));
typedef float v8f __attribute__((ext_vector_type(8)));

#define N_WIRES 10
#define N_LAYERS 4
#define DIM 1024                       // 2^10 amplitudes
#define PADDED(k) ((k) + ((k) >> 4))   // 1 pad word per 16 -> bank-conflict relief
#define PDIM (DIM + (DIM >> 4))        // 1088
#define LAYER_F 1536                   // G0r/G0i | G1r/G1i | GTr/GTi (256 each)

__device__ __forceinline__ void cmul(float ar, float ai, float br, float bi,
                                     float &cr, float &ci) {
  cr = ar * br - ai * bi;
  ci = ar * bi + ai * br;
}

// Composed CNOT-ring basis permutation (control wire w, target wire (w+rr)%10,
// wire w <-> index bit 9-w). Linear over GF(2).
__device__ __forceinline__ int cnot_perm(int m, int rr) {
#pragma unroll
  for (int w = 0; w < N_WIRES; ++w) {
    int t = w + rr; if (t >= N_WIRES) t -= N_WIRES;
    m ^= ((m >> (9 - w)) & 1) << (9 - t);
  }
  return m;
}

// ---------------------------------------------------------------------------
// Kernel A: build fused per-layer group unitaries from weights[4][10][3].
// All three 16x16 operators stored TRANSPOSED (K-major: T[K*16+M]) so WMMA
// A-fragment LDS loads in the main kernel hit 16 consecutive banks.
// ---------------------------------------------------------------------------
__global__ __launch_bounds__(256) void vqc_precompute(const float *__restrict__ W,
                                                      float *__restrict__ G) {
  __shared__ float u[N_WIRES][8]; // per wire: m00r,m00i,m01r,m01i,m10r,m10i,m11r,m11i
  const int tid = threadIdx.x;
  for (int l = 0; l < N_LAYERS; ++l) {
    if (tid < N_WIRES) {
      const float phi   = W[(l * N_WIRES + tid) * 3 + 0];
      const float theta = W[(l * N_WIRES + tid) * 3 + 1];
      const float omega = W[(l * N_WIRES + tid) * 3 + 2];
      const float c = __cosf(0.5f * theta), s = __sinf(0.5f * theta);
      const float a = 0.5f * (phi + omega), b = 0.5f * (phi - omega);
      const float ca = __cosf(a), sa = __sinf(a), cb = __cosf(b), sb = __sinf(b);
      u[tid][0] =  ca * c; u[tid][1] = -sa * c;  // m00 =  e^{-ia} c
      u[tid][2] = -cb * s; u[tid][3] = -sb * s;  // m01 = -e^{+ib} s
      u[tid][4] =  cb * s; u[tid][5] = -sb * s;  // m10 =  e^{-ib} s
      u[tid][6] =  ca * c; u[tid][7] =  sa * c;  // m11 =  e^{+ia} c
    }
    __syncthreads();
    float *L = G + l * LAYER_F;
    const int r = tid >> 4, c = tid & 15;
#pragma unroll
    for (int g = 0; g < 2; ++g) {   // g=0: wires 0..3, g=1: wires 4..7
      float pr = 1.f, pi = 0.f;
#pragma unroll
      for (int q = 0; q < 4; ++q) {
        const int rb = (r >> (3 - q)) & 1, cb2 = (c >> (3 - q)) & 1;
        const float *e = &u[g * 4 + q][(rb * 2 + cb2) * 2];
        float nr, ni; cmul(pr, pi, e[0], e[1], nr, ni);
        pr = nr; pi = ni;
      }
      L[g * 512 + c * 16 + r] = pr;          // transposed store
      L[g * 512 + 256 + c * 16 + r] = pi;
    }
    // GT = I4 (x) (u8 (x) u9): 16x16 block-diagonal on bits 3..0
    {
      const int a = r >> 2, b = r & 3, a2 = c >> 2, b2 = c & 3;
      float pr = 0.f, pi = 0.f;
      if (a == a2) {
        const float *e8 = &u[8][(((b >> 1) * 2) + (b2 >> 1)) * 2];
        const float *e9 = &u[9][(((b & 1) * 2) + (b2 & 1)) * 2];
        cmul(e8[0], e8[1], e9[0], e9[1], pr, pi);
      }
      L[1024 + c * 16 + r] = pr;             // transposed store
      L[1280 + c * 16 + r] = pi;
    }
    __syncthreads();   // u[] reused next layer
  }
}

// ---------------------------------------------------------------------------
// One 16x16 complex group apply.  All pointers are pre-offset by per-lane
// bases; every index below is a compile-time literal after unrolling, so the
// loop body is immediate-offset ds_loads + 2 sign-cndmasks + 2 WMMAs.
//   A  : gate fragment base (K-major), imag at +256 words
//   p1 : first-operand state plane  (real part waves: re; imag waves: im)
//   p2 : second-operand state plane (swapped)
//   dst: output plane (modes 0/1: pre-offset by writeback base;
//        mode 2: plane base, scatter through permBase ^ D(v))
// kkStride/compOff word offsets: g0: 272/68, g1: 17/4, GT: 4/1.
// NO internal barrier: ping-pong buffers make read/write sets disjoint.
// ---------------------------------------------------------------------------
__device__ __forceinline__ void group_apply(const float *__restrict__ A,
                                            const float *p1, const float *p2,
                                            float *dst,
                                            int kkStride, int compOff,
                                            int mode, int rr, int permBase,
                                            bool pimg) {
  v8f acc = {};
#pragma unroll
  for (int kk = 0; kk < 4; ++kk) {
    v2f ar, ai, bF, bS, aS;
    ar.x = A[kk * 64];        ar.y = A[kk * 64 + 16];
    ai.x = A[256 + kk * 64];  ai.y = A[256 + kk * 64 + 16];
    bF.x = p1[kk * kkStride];           bF.y = p1[kk * kkStride + compOff];
    bS.x = p2[kk * kkStride];           bS.y = p2[kk * kkStride + compOff];
    aS.x = pimg ? ai.x : -ai.x;
    aS.y = pimg ? ai.y : -ai.y;
    // part==0 (Dr):  acc += Ar*Br + (-Ai)*Bi
    // part==1 (Di):  acc += Ar*Bi + ( Ai)*Br
    acc = __builtin_amdgcn_wmma_f32_16x16x4_f32(false, ar, false, bF,
                                                (short)0, acc, false, false);
    acc = __builtin_amdgcn_wmma_f32_16x16x4_f32(false, aS, false, bS,
                                                (short)0, acc, false, false);
  }
#pragma unroll
  for (int v = 0; v < 8; ++v) {
    int w;
    if (mode == 0)      w = v * 68;
    else if (mode == 1) w = v * 4 + (v >> 2);
    else                w = PADDED(permBase ^ cnot_perm(v, rr)); // ^literal
    dst[w] = acc[v];
  }
}

// ---------------------------------------------------------------------------
// Kernel B: one block (256 thr = 8 waves) per sample.
// ---------------------------------------------------------------------------
__global__ __launch_bounds__(256) void vqc_kernel(const float *__restrict__ X,
                                                  const float *__restrict__ G,
                                                  const float *__restrict__ bias,
                                                  float *__restrict__ out) {
  __shared__ float SR[2][PDIM];             // ping-pong state planes
  __shared__ float SI[2][PDIM];
  __shared__ float gm[N_LAYERS * LAYER_F];  // all fused gates, staged once
  __shared__ float cs[2 * N_WIRES];
  __shared__ float pLo[32], pHi[32];
  __shared__ int   permT[N_LAYERS][64];     // D_l(c*16) per column c
  __shared__ float red[8];

  const int tid  = threadIdx.x;
  const int lane = tid & 31;
  const int wave = tid >> 5;
  const int s    = blockIdx.x;

  // stage all fused gate matrices (24.6 KB, L2-resident across blocks)
  for (int i = tid; i < N_LAYERS * LAYER_F; i += 256) gm[i] = G[i];

  // CNOT permutation table: one entry per (layer, column).  One-time cost.
  permT[tid >> 6][tid & 63] = cnot_perm((tid & 63) << 4, (tid >> 6) + 1);

  // encoding half-angles: RY(x*pi/2) on |0> -> [cos(x*pi/4), sin(x*pi/4)]
  if (tid < N_WIRES) {
    const float x = X[s * N_WIRES + tid] * 0.7853981633974483f;
    cs[tid] = __cosf(x);
    cs[N_WIRES + tid] = __sinf(x);
  }
  __syncthreads();

  // rank-1 factorization of the product state: bits 9..5 <-> wires 0..4,
  // bits 4..0 <-> wires 5..9
  if (tid < 32) {
    float p = 1.f;
#pragma unroll
    for (int w = 5; w < N_WIRES; ++w)
      p *= cs[((tid >> (9 - w)) & 1) * N_WIRES + w];
    pLo[tid] = p;
  } else if (tid < 64) {
    const int h = tid - 32;
    float p = 1.f;
#pragma unroll
    for (int w = 0; w < 5; ++w)
      p *= cs[((h >> (4 - w)) & 1) * N_WIRES + w];
    pHi[h] = p;
  }
  __syncthreads();

  for (int k = tid; k < DIM; k += 256) {
    SR[0][PADDED(k)] = pHi[k >> 5] * pLo[k & 31];
    SI[0][PADDED(k)] = 0.f;
  }
  __syncthreads();

  // ---- per-lane WMMA constants and hoisted LDS base indices (words) ----
  const int  tile = wave >> 1;          // column tile 0..3 (16 cols each)
  const bool pimg = (wave & 1) != 0;    // false: D.real   true: D.imag
  const int  col  = tile * 16 + (lane & 15);
  const int  kb   = (lane >> 4) << 1;   // WMMA A/B K-base per lane half: 0 or 2
  const int  M    = lane & 15;          // WMMA A row
  const int  laneHi8 = (lane >> 4) << 3;
  const int  cx = col >> 2, cy = col & 3;
  // g=0: PADDED(K0*64+col)      = baseB0 + kk*272 + comp*68
  // g=1: PADDED(x*64+K0*4+y)    = baseB1 + kk*17  + comp*4
  // GT : PADDED(col*16+K0)      = baseB2 + kk*4   + comp
  // g=0 out: PADDED(i*64+col)   = baseW0 + v*68        (i = v + laneHi8)
  // g=1 out: PADDED(x*64+i*4+y) = baseW1 + v*4 + (v>>2)
  const int baseB0 = col + (col >> 4) + kb * 68;
  const int baseB1 = cx * 68 + cy + kb * 4;
  const int baseB2 = col * 17 + kb;
  const int baseW0 = col + (col >> 4) + laneHi8 * 68;
  const int baseW1 = cx * 68 + cy + laneHi8 * 4 + (laneHi8 >> 2);
  const int baseA  = kb * 16 + M;       // A fragment: + kk*64 (+16 for comp)

#pragma unroll
  for (int l = 0; l < N_LAYERS; ++l) {
    const float *Lg = gm + l * LAYER_F;
    const int r0 = (3 * l) & 1;         // compile-time after unroll

    // phase 1: G0 (bits 9..6), buf r0 -> r0^1
    group_apply(Lg + baseA,
                (pimg ? SI[r0] : SR[r0]) + baseB0,
                (pimg ? SR[r0] : SI[r0]) + baseB0,
                (pimg ? SI[r0 ^ 1] : SR[r0 ^ 1]) + baseW0,
                272, 68, 0, 0, 0, pimg);
    __syncthreads();
    // phase 2: G1 (bits 5..2), buf r0^1 -> r0
    group_apply(Lg + 512 + baseA,
                (pimg ? SI[r0 ^ 1] : SR[r0 ^ 1]) + baseB1,
                (pimg ? SR[r0 ^ 1] : SI[r0 ^ 1]) + baseB1,
                (pimg ? SI[r0] : SR[r0]) + baseW1,
                17, 4, 1, 0, 0, pimg);
    __syncthreads();
    // phase 3: GT = I4 (x) G2 (bits 3..0) with CNOT scatter, buf r0 -> r0^1.
    // D(col*16 + v + laneHi8) = permT[l][col] ^ D(laneHi8) ^ D(v)
    {
      const int pb = permT[l][col] ^ ((lane & 16) ? cnot_perm(8, l + 1) : 0);
      group_apply(Lg + 1024 + baseA,
                  (pimg ? SI[r0] : SR[r0]) + baseB2,
                  (pimg ? SR[r0] : SI[r0]) + baseB2,
                  (pimg ? SI[r0 ^ 1] : SR[r0 ^ 1]),   // plane base (scatter)
                  4, 1, 2, l + 1, pb, pimg);
      __syncthreads();
    }
  }

  // after 12 phases the state is back in buffer 0
  // ---- <Z> on wire 9 (index LSB): wave32 shuffle reduction ----
  const float sgn = (tid & 1) ? -1.f : 1.f;
  float acc = 0.f;
#pragma unroll
  for (int j = 0; j < DIM / 256; ++j) {
    const int a = PADDED(tid + 256 * j);
    const float re = SR[0][a], im = SI[0][a];
    acc += re * re + im * im;
  }
  float z = sgn * acc;
#pragma unroll
  for (int m = 16; m > 0; m >>= 1) z += __shfl_xor(z, m, 32);
  if (lane == 0) red[wave] = z;
  __syncthreads();
  if (tid == 0) {
    float t = 0.f;
#pragma unroll
    for (int w = 0; w < 8; ++w) t += red[w];
    out[s] = t + bias[0];
  }
}

// ---------------------------------------------------------------------------
extern "C" void kernel_launch(void *const *d_in, const int *in_sizes, int n_in,
                              void *d_out, int out_size, void *d_ws, size_t ws_size,
                              hipStream_t stream) {
  const float *X = (const float *)d_in[0];       // [B, 10]
  const float *W = (const float *)d_in[1];       // [4, 10, 3]
  const float *bias = (const float *)d_in[2];    // [1]
  float *out = (float *)d_out;                   // [B, 1]
  float *G = (float *)d_ws;                      // 4 * 1536 floats

  const int B = in_sizes[0] / N_WIRES;

  vqc_precompute<<<1, 256, 0, stream>>>(W, G);
  vqc_kernel<<<B, 256, 0, stream>>>(X, G, bias, out);
}